// SkipGramModel_53884659696080
// MI455X (gfx1250) — compile-verified
//
#include <hip/hip_runtime.h>
#include <hip/hip_bf16.h>

// ---- problem constants (from reference) ----
constexpr int V    = 50000;
constexpr int D    = 256;
constexpr int C1   = 2500;
constexpr int C2   = 10000;
constexpr int HEAD = C1 + 2;      // 2502
constexpr int H1   = D / 4;       // 64
constexpr int H2   = D / 16;      // 16
constexpr int NTOK = 4096;

// kernel modes
constexpr int MODE_STORE = 0;     // write logits tile to Y
constexpr int MODE_LSE   = 1;     // exp-sum + target pick
constexpr int MODE_HEAD  = 2;     // exp-sum + target pick + 2 fixed-column picks

typedef float v2f __attribute__((ext_vector_type(2)));
typedef float v8f __attribute__((ext_vector_type(8)));

__device__ __forceinline__ v8f wmma_f32_k4(v2f a, v2f b, v8f c) {
    // D = A(16x4, f32) * B(4x16, f32) + C(16x16, f32)
    return __builtin_amdgcn_wmma_f32_16x16x4_f32(
        /*neg_a=*/false, a, /*neg_b=*/false, b,
        /*c_mod=*/(short)0, c, /*reuse_a=*/false, /*reuse_b=*/false);
}

// ---------------------------------------------------------------------------
__global__ void init_zero_k(float* __restrict__ p, int n) {
    int i = blockIdx.x * blockDim.x + threadIdx.x;
    if (i < n) p[i] = 0.0f;
}

// ---------------------------------------------------------------------------
// h[i][:] = emb[x[i]][:]   (float4 vectorized gather)
__global__ void gather_k(const float* __restrict__ emb, const int* __restrict__ x,
                         float* __restrict__ h) {
    int t  = blockIdx.x * blockDim.x + threadIdx.x;   // over NTOK * (D/4)
    int i  = t >> 6;                                  // D/4 == 64
    int d4 = t & 63;
    const float4* src = reinterpret_cast<const float4*>(emb + (long)x[i] * D) + d4;
    float4*       dst = reinterpret_cast<float4*>(h + (long)i * D) + d4;
    *dst = *src;
}

// ---------------------------------------------------------------------------
// Unified WMMA GEMM:  logits = A[NTOK,K] @ W[O,K]^T
//  MODE_STORE: Y[row, col] = logit                       (out0 = Y, O mult of 16)
//  MODE_LSE  : accSum[row] += sum(exp(logit over cols)), accPick[row] = logit at
//              clip(target-sub, 0, clampMax)             (out0 = accSum)
//  MODE_HEAD : MODE_LSE + picks at f0col / f1col
// Block = 4 waves; each wave owns 16 rows. All waves share column tiles, which
// are cooperatively staged into a double-buffered LDS tile. A tiles are staged
// wave-privately in LDS. Columns split over gridDim.y chunks (atomicAdd merge).
template <int K, int MODE>
__global__ void __launch_bounds__(128)
gemm_k(const float* __restrict__ A, const float* __restrict__ W,
       const int* __restrict__ target, int O, int sub, int clampMax,
       float* __restrict__ out0, float* __restrict__ accPick,
       float* __restrict__ accF0, float* __restrict__ accF1,
       int f0col, int f1col) {
    constexpr int KT  = K / 4;         // WMMAs (k-steps of 4) per 16x16 tile
    constexpr int SK  = K + 4;         // padded LDS row stride (conflict-free)
    constexpr int KF4 = 16 * (K / 4);  // float4s per 16-row tile
    __shared__ float sA[4 * 16 * SK];
    __shared__ float sB[2 * 16 * SK];

    int lane = threadIdx.x & 31;
    int wv   = threadIdx.x >> 5;
    int rowbase = (blockIdx.x * 4 + wv) * 16;
    int m    = lane & 15;
    int half = lane >> 4;
    int koff = half * 2;

    // ---- stage wave-private A tile: 16 rows x K, padded stride ----
    for (int i = lane; i < KF4; i += 32) {
        int r = i / (K / 4), c4 = i % (K / 4);
        *reinterpret_cast<float4*>(&sA[(wv * 16 + r) * SK + c4 * 4]) =
            *reinterpret_cast<const float4*>(A + (long)(rowbase + r) * K + c4 * 4);
    }

    // ---- per-lane target columns for its 8 rows (LSE modes only) ----
    int tc[8];
    float sacc[8], pacc[8], f0a[8], f1a[8];
    if constexpr (MODE != MODE_STORE) {
#pragma unroll
        for (int v = 0; v < 8; ++v) {
            int t = target[rowbase + half * 8 + v] - sub;
            t = t < 0 ? 0 : t;
            tc[v] = t > clampMax ? clampMax : t;
            sacc[v] = 0.f; pacc[v] = 0.f; f0a[v] = 0.f; f1a[v] = 0.f;
        }
    }

    int Ot     = (O + 15) >> 4;
    int chunks = gridDim.y;
    int tpc    = (Ot + chunks - 1) / chunks;
    int t0     = blockIdx.y * tpc;
    int t1     = t0 + tpc; if (t1 > Ot) t1 = Ot;
    // (launch configs guarantee t0 < t1 for every block)

    // cooperative B-tile copy: 16 weight rows x K floats into buffer `buf`
    auto copyB = [&](int tile, int buf) {
        int colbase = tile * 16;
        for (int i = threadIdx.x; i < KF4; i += 128) {
            int r = i / (K / 4), c4 = i % (K / 4);
            int wr = colbase + r; wr = wr < O ? wr : O - 1;   // clamp OOB rows
            *reinterpret_cast<float4*>(&sB[(buf * 16 + r) * SK + c4 * 4]) =
                *reinterpret_cast<const float4*>(W + (long)wr * K + c4 * 4);
        }
    };

    copyB(t0, 0);
    __syncthreads();

    const float* sAp = sA + (wv * 16 + m) * SK + koff;

    for (int t = t0; t < t1; ++t) {
        int cur = (t - t0) & 1;
        if (t + 1 < t1) copyB(t + 1, cur ^ 1);   // prefetch next tile

        int colbase = t * 16;
        int col     = colbase + m;
        const float* sBp = sB + (cur * 16 + m) * SK + koff;

        v8f c = {};
        // 1-deep software pipeline over LDS b64 loads
        v2f a0 = *reinterpret_cast<const v2f*>(sAp);
        v2f b0 = *reinterpret_cast<const v2f*>(sBp);
#pragma unroll
        for (int kt = 0; kt < KT; ++kt) {
            v2f a1, b1;
            if (kt + 1 < KT) {
                a1 = *reinterpret_cast<const v2f*>(sAp + (kt + 1) * 4);
                b1 = *reinterpret_cast<const v2f*>(sBp + (kt + 1) * 4);
            }
            c = wmma_f32_k4(a0, b0, c);
            a0 = a1; b0 = b1;
        }

        if constexpr (MODE == MODE_STORE) {
            int rl = half * 8;
#pragma unroll
            for (int v = 0; v < 8; ++v)
                out0[(long)(rowbase + rl + v) * O + col] = c[v];
        } else {
            bool valid = col < O;
#pragma unroll
            for (int v = 0; v < 8; ++v) {
                float val = c[v];
                sacc[v] += valid ? __expf(val) : 0.f;
                if (col == tc[v]) pacc[v] = val;      // tc < O, so match implies valid
                if constexpr (MODE == MODE_HEAD) {
                    if (col == f0col) f0a[v] = val;
                    if (col == f1col) f1a[v] = val;
                }
            }
        }
        __syncthreads();   // cur fully consumed; next buffer fully written
    }

    if constexpr (MODE != MODE_STORE) {
        // ---- reduce across the 16 lanes of each half-wave ----
#pragma unroll
        for (int v = 0; v < 8; ++v) {
            float s = sacc[v], p = pacc[v], a0 = f0a[v], a1 = f1a[v];
            for (int off = 1; off < 16; off <<= 1) {
                s  += __shfl_xor(s,  off, 32);
                p  += __shfl_xor(p,  off, 32);
                if constexpr (MODE == MODE_HEAD) {
                    a0 += __shfl_xor(a0, off, 32);
                    a1 += __shfl_xor(a1, off, 32);
                }
            }
            sacc[v] = s; pacc[v] = p; f0a[v] = a0; f1a[v] = a1;
        }
        if (m == 0) {
#pragma unroll
            for (int v = 0; v < 8; ++v) {
                int r = rowbase + half * 8 + v;
                atomicAdd(&out0[r],    sacc[v]);
                atomicAdd(&accPick[r], pacc[v]);
                if constexpr (MODE == MODE_HEAD) {
                    atomicAdd(&accF0[r], f0a[v]);
                    atomicAdd(&accF1[r], f1a[v]);
                }
            }
        }
    }
}

// ---------------------------------------------------------------------------
__global__ void combine_k(const int* __restrict__ target,
                          const float* __restrict__ sumH, const float* __restrict__ pickH,
                          const float* __restrict__ f0H,  const float* __restrict__ f1H,
                          const float* __restrict__ sum1, const float* __restrict__ pick1,
                          const float* __restrict__ sum2, const float* __restrict__ pick2,
                          float* __restrict__ out) {
    int i = blockIdx.x * blockDim.x + threadIdx.x;
    if (i >= NTOK) return;
    float lseH = __logf(sumH[i]);
    int t = target[i];
    float o;
    if (t < C1)      o = pickH[i] - lseH;
    else if (t < C2) o = (f0H[i] - lseH) + (pick1[i] - __logf(sum1[i]));
    else             o = (f1H[i] - lseH) + (pick2[i] - __logf(sum2[i]));
    out[i] = o;
}

__global__ void loss_k(const float* __restrict__ out, float* __restrict__ loss_out) {
    __shared__ float red[8];
    float s = 0.f;
    for (int i = threadIdx.x; i < NTOK; i += 256) s += out[i];
    for (int off = 16; off >= 1; off >>= 1) s += __shfl_xor(s, off, 32);
    if ((threadIdx.x & 31) == 0) red[threadIdx.x >> 5] = s;
    __syncthreads();
    if (threadIdx.x == 0) {
        float tot = 0.f;
        for (int w = 0; w < 8; ++w) tot += red[w];
        loss_out[0] = -tot / (float)NTOK;
    }
}

// ---------------------------------------------------------------------------
extern "C" void kernel_launch(void* const* d_in, const int* in_sizes, int n_in,
                              void* d_out, int out_size, void* d_ws, size_t ws_size,
                              hipStream_t stream) {
    (void)in_sizes; (void)n_in; (void)out_size; (void)ws_size;
    const int*   x      = (const int*)d_in[0];
    const int*   target = (const int*)d_in[1];
    const float* emb    = (const float*)d_in[2];
    const float* Wh     = (const float*)d_in[3];
    const float* P1     = (const float*)d_in[4];
    const float* W1     = (const float*)d_in[5];
    const float* P2     = (const float*)d_in[6];
    const float* W2     = (const float*)d_in[7];
    float* out = (float*)d_out;

    // workspace layout
    float* h    = (float*)d_ws;                 // NTOK*D
    float* hp1  = h   + (long)NTOK * D;         // NTOK*H1
    float* hp2  = hp1 + (long)NTOK * H1;        // NTOK*H2
    float* acc  = hp2 + (long)NTOK * H2;        // 8 * NTOK accumulators
    float* sumH = acc + 0 * NTOK;
    float* pickH= acc + 1 * NTOK;
    float* f0H  = acc + 2 * NTOK;
    float* f1H  = acc + 3 * NTOK;
    float* sum1 = acc + 4 * NTOK;
    float* pick1= acc + 5 * NTOK;
    float* sum2 = acc + 6 * NTOK;
    float* pick2= acc + 7 * NTOK;

    // 1) zero accumulators
    init_zero_k<<<(8 * NTOK + 255) / 256, 256, 0, stream>>>(acc, 8 * NTOK);

    // 2) embedding gather
    gather_k<<<(NTOK * (D / 4)) / 256, 256, 0, stream>>>(emb, x, h);

    // 3) tail projections (WMMA store-GEMMs, LDS-pipelined)
    gemm_k<256, MODE_STORE><<<dim3(NTOK / 64, 1), 128, 0, stream>>>(
        h, P1, nullptr, H1, 0, 0, hp1, nullptr, nullptr, nullptr, -1, -1);
    gemm_k<256, MODE_STORE><<<dim3(NTOK / 64, 1), 128, 0, stream>>>(
        h, P2, nullptr, H2, 0, 0, hp2, nullptr, nullptr, nullptr, -1, -1);

    // 4) fused GEMM + expsum/pick for each cluster
    //    head: Ot=157, 4 chunks -> 256 blocks; tail1: Ot=469, 8 chunks -> 512;
    //    tail2: Ot=2500, 16 chunks -> 1024 blocks. All chunks non-empty.
    gemm_k<256, MODE_HEAD><<<dim3(NTOK / 64, 4),  128, 0, stream>>>(
        h,   Wh, target, HEAD,    0, C1 - 1,       sumH, pickH, f0H, f1H, C1, C1 + 1);
    gemm_k<64,  MODE_LSE ><<<dim3(NTOK / 64, 8),  128, 0, stream>>>(
        hp1, W1, target, C2 - C1, C1, C2 - C1 - 1, sum1, pick1, nullptr, nullptr, -1, -1);
    gemm_k<16,  MODE_LSE ><<<dim3(NTOK / 64, 16), 128, 0, stream>>>(
        hp2, W2, target, V - C2,  C2, V - C2 - 1,  sum2, pick2, nullptr, nullptr, -1, -1);

    // 5) select per-token log-prob, then loss
    combine_k<<<NTOK / 256, 256, 0, stream>>>(target, sumH, pickH, f0H, f1H,
                                              sum1, pick1, sum2, pick2, out);
    loss_k<<<1, 256, 0, stream>>>(out, out + NTOK);
}